// RELEncoder_10522669875603
// MI455X (gfx1250) — compile-verified
//
#include <hip/hip_runtime.h>
#include <math.h>

typedef __attribute__((ext_vector_type(2))) float v2f;
typedef __attribute__((ext_vector_type(8))) float v8f;

#define BB    4
#define NNODE 100
#define EEDGE 9900
#define NINP  256
#define HH    256
#define RRR   64
#define SSS   4
#define BN_EPS 1e-5f

// LDS weight panel: 32 K-pair rows, each 128 data dwords (64 cols x 2 interleaved K)
// + 32 pad dwords so the two wave-halves (pair p vs p+1) land on disjoint bank sets.
#define PSTR 160

__device__ __forceinline__ float elu1(float v) { return v > 0.f ? v : (__expf(v) - 1.f); }
__device__ __forceinline__ float sigm(float v) { return 1.f / (1.f + __expf(-v)); }

__device__ __forceinline__ v8f wmma4(v2f a, v2f b, v8f c) {
  return __builtin_amdgcn_wmma_f32_16x16x4_f32(false, a, false, b, (short)0, c, false, false);
}

// ---------------- index extraction from one-hot matrices ----------------
__global__ void kidx(const float* __restrict__ rel_rec, const float* __restrict__ rel_send,
                     int* __restrict__ rec, int* __restrict__ snd) {
  int e = blockIdx.x * blockDim.x + threadIdx.x;
  if (e >= EEDGE) return;
  int rb = 0, sb = 0;
  for (int n = 0; n < NNODE; ++n) {
    if (rel_rec[e * NNODE + n] > 0.5f) rb = n;
    if (rel_send[e * NNODE + n] > 0.5f) sb = n;
  }
  rec[e] = rb;
  snd[e] = sb;
}

// stage 64(K) x 64(N) chunk of W into K-pair-interleaved LDS:
// ldsW[p*PSTR + n*2 + q] = W[(kc+2p+q)*N + cbase + n]
__device__ __forceinline__ void stageW(float* ldsW, const float* __restrict__ W,
                                       int kc, int cbase, int N, int tid) {
  for (int i = 0; i < 2; ++i) {
    int idx = tid + (i << 8);            // 0..511: 32 pairs x 16 col-quads
    int p  = idx >> 4;
    int c4 = (idx & 15) << 2;
    const float* wr0 = W + (size_t)(kc + 2 * p) * N + cbase + c4;
    float4 r0 = *(const float4*)wr0;
    float4 r1 = *(const float4*)(wr0 + N);
    float* d = &ldsW[p * PSTR + (c4 << 1)];
    *(float4*)(d)     = make_float4(r0.x, r1.x, r0.y, r1.y);
    *(float4*)(d + 4) = make_float4(r0.z, r1.z, r0.w, r1.w);
  }
}

// ---------------- fp32 WMMA GEMM with LDS weight staging ----------------
// out = ELU(X[M,K] @ W[K,N] + b). Block = 8 waves sharing one 64-col strip.
// M%16==0, N%64==0, K%64==0.
__global__ void kgemm_elu(const float* __restrict__ X, int ldx,
                          const float* __restrict__ W, const float* __restrict__ bias,
                          float* __restrict__ out, int M, int K, int N) {
  __shared__ float ldsW[32 * PSTR];

  int colStrips = N >> 6;
  int rowTiles  = M >> 4;
  int cs = blockIdx.x % colStrips;
  int rb = blockIdx.x / colStrips;
  int wid  = threadIdx.x >> 5;
  int lane = threadIdx.x & 31;
  int rt = rb * 8 + wid;
  bool active = (rt < rowTiles);        // wave-uniform

  int m    = lane & 15;
  int half = lane >> 4;
  int row0 = rt << 4;
  int cbase = cs << 6;

  v8f c0 = {}, c1 = {}, c2 = {}, c3 = {};
  const float* xr = X + (size_t)(row0 + m) * ldx;

  for (int kc = 0; kc < K; kc += 64) {
    __syncthreads();
    stageW(ldsW, W, kc, cbase, N, threadIdx.x);
    __syncthreads();
    if (active) {
      for (int kl = 0; kl < 64; kl += 4) {
        // A: lanes 0-15 carry K0/K1, lanes 16-31 carry K2/K3
        float2 af = *(const float2*)(xr + kc + kl + 2 * half);
        v2f a; a.x = af.x; a.y = af.y;
        const float* wp = &ldsW[((kl >> 1) + half) * PSTR];
        v2f b;
        b = *(const v2f*)(wp + (m << 1));          c0 = wmma4(a, b, c0);
        b = *(const v2f*)(wp + ((m + 16) << 1));   c1 = wmma4(a, b, c1);
        b = *(const v2f*)(wp + ((m + 32) << 1));   c2 = wmma4(a, b, c2);
        b = *(const v2f*)(wp + ((m + 48) << 1));   c3 = wmma4(a, b, c3);
      }
    }
  }
  if (!active) return;

  int n0 = cbase + m;
  for (int i = 0; i < 8; ++i) {          // C: VGPR i = rows i / i+8, col = lane&15
    int r = row0 + i + 8 * half;
    float* orow = out + (size_t)r * N;
    orow[n0]      = elu1(c0[i] + bias[n0]);
    orow[n0 + 16] = elu1(c1[i] + bias[n0 + 16]);
    orow[n0 + 32] = elu1(c2[i] + bias[n0 + 32]);
    orow[n0 + 48] = elu1(c3[i] + bias[n0 + 48]);
  }
}

// ---------------- BatchNorm stats ----------------
__global__ void kbn_zero(float* sum, float* sumsq) {
  int c = threadIdx.x;
  sum[c] = 0.f;
  sumsq[c] = 0.f;
}

__global__ void kbn_part(const float* __restrict__ X, int M,
                         float* __restrict__ sum, float* __restrict__ sumsq) {
  int c = threadIdx.x;   // one column per thread, HH==256==blockDim
  float s = 0.f, s2 = 0.f;
  for (int r = blockIdx.x; r < M; r += gridDim.x) {
    float v = X[(size_t)r * HH + c];
    s += v;
    s2 += v * v;
  }
  atomicAdd(&sum[c], s);
  atomicAdd(&sumsq[c], s2);
}

__global__ void kbn_fin(const float* __restrict__ sum, const float* __restrict__ sumsq,
                        float invM, float* __restrict__ mu, float* __restrict__ rsig) {
  int c = threadIdx.x;
  float mval = sum[c] * invM;
  mu[c] = mval;
  float var = sumsq[c] * invM - mval * mval;
  rsig[c] = rsqrtf(var + BN_EPS);
}

// out[i*ostride + ooff] = (X[i]-mu)*rsig*g + beta   (ostride=1 contiguous, =S for [.,H,S] layout)
__global__ void kbn_apply(const float* __restrict__ X, const float* __restrict__ mu,
                          const float* __restrict__ rsig, const float* __restrict__ g,
                          const float* __restrict__ beta, float* __restrict__ out,
                          int M, int ostride, int ooff) {
  int i = blockIdx.x * blockDim.x + threadIdx.x;
  if (i >= M * HH) return;
  int c = i & (HH - 1);
  float v = (X[i] - mu[c]) * rsig[c] * g[c] + beta[c];
  out[(size_t)i * ostride + ooff] = v;
}

// ---------------- ht2rel = tp[rec] - hp[snd], write contiguous + strided output ----------------
__global__ void kht(const float* __restrict__ tp, const float* __restrict__ hp,
                    const int* __restrict__ rec, const int* __restrict__ snd,
                    float* __restrict__ bufHT, float* __restrict__ outHT, int j) {
  int i = blockIdx.x * blockDim.x + threadIdx.x;
  if (i >= BB * EEDGE * HH) return;
  int h = i & (HH - 1);
  int row = i >> 8;              // /HH
  int b = row / EEDGE;
  int e = row - b * EEDGE;
  float v = tp[(size_t)(b * NNODE + rec[e]) * HH + h] - hp[(size_t)(b * NNODE + snd[e]) * HH + h];
  bufHT[i] = v;
  outHT[(size_t)i * SSS + j] = v;
}

// ---------------- readout: sigmoid([HT|U] @ roW + rob), col0 -> sigmoid(-inf)=0 ----------------
// K=512 (first 256 from HT, next 256 from U), N=64. 8 waves/block, W staged in LDS.
__global__ void kreadout(const float* __restrict__ HT, const float* __restrict__ U,
                         const float* __restrict__ Wro, const float* __restrict__ bro,
                         float* __restrict__ outE, int j) {
  __shared__ float ldsW[32 * PSTR];

  int rowTiles = (BB * EEDGE) >> 4;   // 2475
  int wid  = threadIdx.x >> 5;
  int lane = threadIdx.x & 31;
  int rt = blockIdx.x * 8 + wid;
  bool active = (rt < rowTiles);      // wave-uniform

  int m    = lane & 15;
  int half = lane >> 4;
  int row0 = rt << 4;

  v8f c0 = {}, c1 = {}, c2 = {}, c3 = {};
  const float* ht = HT + (size_t)(row0 + m) * HH;
  const float* uu = U + (size_t)(row0 + m) * HH;

  for (int kc = 0; kc < 2 * HH; kc += 64) {
    __syncthreads();
    stageW(ldsW, Wro, kc, 0, RRR, threadIdx.x);
    __syncthreads();
    if (active) {
      const float* src = (kc < HH) ? (ht + kc) : (uu + kc - HH);
      for (int kl = 0; kl < 64; kl += 4) {
        float2 af = *(const float2*)(src + kl + 2 * half);
        v2f a; a.x = af.x; a.y = af.y;
        const float* wp = &ldsW[((kl >> 1) + half) * PSTR];
        v2f b;
        b = *(const v2f*)(wp + (m << 1));          c0 = wmma4(a, b, c0);
        b = *(const v2f*)(wp + ((m + 16) << 1));   c1 = wmma4(a, b, c1);
        b = *(const v2f*)(wp + ((m + 32) << 1));   c2 = wmma4(a, b, c2);
        b = *(const v2f*)(wp + ((m + 48) << 1));   c3 = wmma4(a, b, c3);
      }
    }
  }
  if (!active) return;

  for (int i = 0; i < 8; ++i) {
    int r = row0 + i + 8 * half;
    size_t base = ((size_t)r * RRR) * SSS + j;
    float v0 = (m == 0) ? 0.f : sigm(c0[i] + bro[m]);   // column 0 forced to -inf -> 0
    outE[base + (size_t)m * SSS] = v0;
    outE[base + (size_t)(m + 16) * SSS] = sigm(c1[i] + bro[m + 16]);
    outE[base + (size_t)(m + 32) * SSS] = sigm(c2[i] + bro[m + 32]);
    outE[base + (size_t)(m + 48) * SSS] = sigm(c3[i] + bro[m + 48]);
  }
}

// ---------------- host-side helpers ----------------
static void gemm(hipStream_t s, const float* X, int ldx, const float* W, const float* b,
                 float* out, int M, int K, int N) {
  int rowTiles = M >> 4;
  int colStrips = N >> 6;
  int rowBlocks = (rowTiles + 7) / 8;
  kgemm_elu<<<rowBlocks * colStrips, 256, 0, s>>>(X, ldx, W, b, out, M, K, N);
}

static void bn_stats(hipStream_t s, const float* X, int M,
                     float* sum, float* sumsq, float* mu, float* rsig) {
  kbn_zero<<<1, 256, 0, s>>>(sum, sumsq);
  int nb = (M > 1024) ? 64 : 2;
  kbn_part<<<nb, 256, 0, s>>>(X, M, sum, sumsq);
  kbn_fin<<<1, 256, 0, s>>>(sum, sumsq, 1.0f / (float)M, mu, rsig);
}

extern "C" void kernel_launch(void* const* d_in, const int* in_sizes, int n_in,
                              void* d_out, int out_size, void* d_ws, size_t ws_size,
                              hipStream_t stream) {
  const float* node     = (const float*)d_in[0];
  const float* edge     = (const float*)d_in[1];
  const float* rels     = (const float*)d_in[2];
  const float* rel_rec  = (const float*)d_in[3];
  const float* rel_send = (const float*)d_in[4];
  const float* hW1 = (const float*)d_in[5];  const float* hb1 = (const float*)d_in[6];
  const float* hW2 = (const float*)d_in[7];  const float* hb2 = (const float*)d_in[8];
  const float* hg  = (const float*)d_in[9];  const float* hbe = (const float*)d_in[10];
  const float* tW1 = (const float*)d_in[11]; const float* tb1 = (const float*)d_in[12];
  const float* tW2 = (const float*)d_in[13]; const float* tb2 = (const float*)d_in[14];
  const float* tg  = (const float*)d_in[15]; const float* tbe = (const float*)d_in[16];
  const float* uW1 = (const float*)d_in[17]; const float* ub1 = (const float*)d_in[18];
  const float* uW2 = (const float*)d_in[19]; const float* ub2 = (const float*)d_in[20];
  const float* ug  = (const float*)d_in[21]; const float* ube = (const float*)d_in[22];
  const float* mW1 = (const float*)d_in[23]; const float* mb1 = (const float*)d_in[24];
  const float* mW2 = (const float*)d_in[25]; const float* mb2 = (const float*)d_in[26];
  const float* mg  = (const float*)d_in[27]; const float* mbe = (const float*)d_in[28];
  const float* pW1 = (const float*)d_in[29]; const float* pb1 = (const float*)d_in[30];
  const float* pW2 = (const float*)d_in[31]; const float* pb2 = (const float*)d_in[32];
  const float* pg  = (const float*)d_in[33]; const float* pbe = (const float*)d_in[34];
  const float* roW = (const float*)d_in[35]; const float* rob = (const float*)d_in[36];

  float* dout = (float*)d_out;
  const size_t o_ht  = (size_t)BB * EEDGE * RRR * SSS;          // e_out size
  const size_t o_rel = o_ht + (size_t)BB * EEDGE * HH * SSS;    // + ht2rels size

  // workspace layout (floats)
  float* wsf = (float*)d_ws;
  int* rec = (int*)wsf;
  int* snd = (int*)(wsf + EEDGE);
  float* sum   = wsf + 2 * EEDGE;
  float* sumsq = sum + HH;
  float* mu    = sumsq + HH;
  float* rsig  = mu + HH;
  const size_t SB = (size_t)BB * NNODE * HH;      // 102400
  float* s0 = rsig + HH;          // scratch layer-1 (node-side)
  float* s1 = s0 + SB;            // head_n
  float* s2 = s1 + SB;            // tail_n
  float* s3 = s2 + SB;            // scratch pre-BN h2 (node-side)
  float* s4 = s3 + SB;            // hp_n
  float* s5 = s4 + SB;            // tp_n
  const size_t BIG = (size_t)BB * EEDGE * HH;     // 10,137,600
  float* bufA  = s5 + SB;
  float* bufB  = bufA + BIG;
  float* bufU  = bufB + BIG;
  float* bufHT = bufU + BIG;

  const int MN = BB * NNODE;    // 400
  const int ME = BB * EEDGE;    // 39600

  // one-hot -> indices
  kidx<<<(EEDGE + 255) / 256, 256, 0, stream>>>(rel_rec, rel_send, rec, snd);

  // head = mlp(node)
  gemm(stream, node, NINP, hW1, hb1, s0, MN, NINP, HH);
  gemm(stream, s0, HH, hW2, hb2, s3, MN, HH, HH);
  bn_stats(stream, s3, MN, sum, sumsq, mu, rsig);
  kbn_apply<<<(MN * HH + 255) / 256, 256, 0, stream>>>(s3, mu, rsig, hg, hbe, s1, MN, 1, 0);

  // tail = mlp(node)
  gemm(stream, node, NINP, tW1, tb1, s0, MN, NINP, HH);
  gemm(stream, s0, HH, tW2, tb2, s3, MN, HH, HH);
  bn_stats(stream, s3, MN, sum, sumsq, mu, rsig);
  kbn_apply<<<(MN * HH + 255) / 256, 256, 0, stream>>>(s3, mu, rsig, tg, tbe, s2, MN, 1, 0);

  // union = mlp(edge)
  gemm(stream, edge, NINP, uW1, ub1, bufA, ME, NINP, HH);
  gemm(stream, bufA, HH, uW2, ub2, bufB, ME, HH, HH);
  bn_stats(stream, bufB, ME, sum, sumsq, mu, rsig);
  kbn_apply<<<(ME * HH + 255) / 256, 256, 0, stream>>>(bufB, mu, rsig, ug, ube, bufU, ME, 1, 0);

  for (int j = 0; j < SSS; ++j) {
    const float* mW1j = mW1 + (size_t)j * HH * HH;
    const float* mb1j = mb1 + (size_t)j * HH;
    const float* mW2j = mW2 + (size_t)j * HH * HH;
    const float* mb2j = mb2 + (size_t)j * HH;
    const float* mgj  = mg  + (size_t)j * HH;
    const float* mbej = mbe + (size_t)j * HH;
    const float* pW1j = pW1 + (size_t)j * RRR * HH;
    const float* pb1j = pb1 + (size_t)j * HH;
    const float* pW2j = pW2 + (size_t)j * HH * HH;
    const float* pb2j = pb2 + (size_t)j * HH;
    const float* pgj  = pg  + (size_t)j * HH;
    const float* pbej = pbe + (size_t)j * HH;
    const float* roWj = roW + (size_t)j * 2 * HH * RRR;
    const float* robj = rob + (size_t)j * RRR;

    // hp = mlp(head)
    gemm(stream, s1, HH, mW1j, mb1j, s0, MN, HH, HH);
    gemm(stream, s0, HH, mW2j, mb2j, s3, MN, HH, HH);
    bn_stats(stream, s3, MN, sum, sumsq, mu, rsig);
    kbn_apply<<<(MN * HH + 255) / 256, 256, 0, stream>>>(s3, mu, rsig, mgj, mbej, s4, MN, 1, 0);

    // tp = mlp(tail)
    gemm(stream, s2, HH, mW1j, mb1j, s0, MN, HH, HH);
    gemm(stream, s0, HH, mW2j, mb2j, s3, MN, HH, HH);
    bn_stats(stream, s3, MN, sum, sumsq, mu, rsig);
    kbn_apply<<<(MN * HH + 255) / 256, 256, 0, stream>>>(s3, mu, rsig, mgj, mbej, s5, MN, 1, 0);

    // r2r = mlp(rels)  -> normalized straight into rel2rels output [B,E,H,S]
    gemm(stream, rels, RRR, pW1j, pb1j, bufA, ME, RRR, HH);
    gemm(stream, bufA, HH, pW2j, pb2j, bufB, ME, HH, HH);
    bn_stats(stream, bufB, ME, sum, sumsq, mu, rsig);
    kbn_apply<<<(ME * HH + 255) / 256, 256, 0, stream>>>(bufB, mu, rsig, pgj, pbej,
                                                         dout + o_rel, ME, SSS, j);

    // ht2rel = tp[rec] - hp[snd]  -> bufHT + ht2rels output [B,E,H,S]
    kht<<<(ME * HH + 255) / 256, 256, 0, stream>>>(s5, s4, rec, snd, bufHT, dout + o_ht, j);

    // e_out[..., j] = sigmoid([ht|union] @ roW_j + rob_j), col0 = 0
    int rowTiles = ME >> 4;
    kreadout<<<(rowTiles + 7) / 8, 256, 0, stream>>>(bufHT, bufU, roWj, robj, dout, j);
  }
}